// QuantumCircuitLayer_67405216743955
// MI455X (gfx1250) — compile-verified
//
#include <hip/hip_runtime.h>

#ifndef __has_builtin
#define __has_builtin(x) 0
#endif

#define NQ  16
#define NL  8
#define TPB 256

#if __has_builtin(__builtin_amdgcn_global_load_async_to_lds_b128)
#define HAVE_ASYNC_LDS 1
#else
#define HAVE_ASYNC_LDS 0
#endif

typedef __attribute__((ext_vector_type(4))) int v4i;
typedef __attribute__((address_space(1))) v4i global_v4i;   // global AS
typedef __attribute__((address_space(3))) v4i local_v4i;    // LDS AS

__device__ __forceinline__ void wait_async0() {
#if __has_builtin(__builtin_amdgcn_s_wait_asynccnt)
  __builtin_amdgcn_s_wait_asynccnt(0);
#else
  asm volatile("s_wait_asynccnt 0" ::: "memory");
#endif
}

// ---------------------------------------------------------------------------
// Prep kernel (1 block): fold Rx(t1),Ry(t2),Rz(t3) — all planar rotations on
// (r,i) — into one rotation by phi = 0.5*(ry+rz-rx); also precompute sigmoids.
// ws layout (floats): [0,256)  = {cos,sin} per (layer,qubit)
//                     [256,376)= sigmoid(ent) per (layer, qubit-1)
// ---------------------------------------------------------------------------
__global__ void qc_prep(const float* __restrict__ rot,
                        const float* __restrict__ ent,
                        float* __restrict__ ws) {
  const int t = threadIdx.x;
  if (t < NL * NQ) {
    float rx = rot[t * 3 + 0];
    float ry = rot[t * 3 + 1];
    float rz = rot[t * 3 + 2];
    float phi = 0.5f * (ry + rz - rx);
    float rev = phi * 0.15915494309189535f;      // v_cos/v_sin take revolutions
    ws[2 * t + 0] = __builtin_amdgcn_cosf(rev);
    ws[2 * t + 1] = __builtin_amdgcn_sinf(rev);
  }
  if (t < NL * (NQ - 1)) {
    float e = ent[t];
    float z = __builtin_amdgcn_exp2f(-e * 1.4426950408889634f);   // exp(-e)
    ws[2 * NL * NQ + t] = __builtin_amdgcn_rcpf(1.0f + z);        // sigmoid
  }
}

// ---------------------------------------------------------------------------
// Main kernel: one batch element per lane, full 16-qubit complex state in
// VGPRs. Input staged via gfx1250 async global->LDS (own-wave data only, so
// s_wait_asynccnt suffices, no workgroup barrier).
// ---------------------------------------------------------------------------
__global__ __launch_bounds__(TPB) void qc_main(const float* __restrict__ x,
                                               const float* __restrict__ ws,
                                               float* __restrict__ out,
                                               int batch) {
#if HAVE_ASYNC_LDS
  __shared__ float tile[TPB * NQ];   // 16 KB staging tile
#endif
  const int t = threadIdx.x;
  const long long blk = (long long)blockIdx.x * TPB;
  const int idx = (int)(blk + t);
  if (idx >= batch) return;

  float xr[NQ];
#if HAVE_ASYNC_LDS
  {
    const float* gp = x + (long long)idx * NQ;
    #pragma unroll
    for (int k = 0; k < 4; ++k) {
      __builtin_amdgcn_global_load_async_to_lds_b128(
          (global_v4i*)(gp + 4 * k),
          (local_v4i*)(&tile[t * NQ + 4 * k]),
          0, 0);
    }
    wait_async0();
    #pragma unroll
    for (int q = 0; q < NQ; ++q) xr[q] = tile[t * NQ + q];
  }
#else
  {
    const float4* xv = (const float4*)(x + (size_t)idx * NQ);
    #pragma unroll
    for (int k = 0; k < 4; ++k) {
      float4 v = xv[k];
      xr[4 * k + 0] = v.x; xr[4 * k + 1] = v.y;
      xr[4 * k + 2] = v.z; xr[4 * k + 3] = v.w;
    }
  }
#endif

  // init: r = cos(x*pi/2) = v_cos(x/4), i = sin(x*pi/2) = v_sin(x/4)
  float r[NQ], im[NQ];
  #pragma unroll
  for (int q = 0; q < NQ; ++q) {
    float rev = xr[q] * 0.25f;
    r[q]  = __builtin_amdgcn_cosf(rev);
    im[q] = __builtin_amdgcn_sinf(rev);
  }

  #pragma unroll 1
  for (int L = 0; L < NL; ++L) {
    // folded rotation: (r,i) <- Rot(phi) * (r,i)
    const float* cs = ws + L * (2 * NQ);
    #pragma unroll
    for (int q = 0; q < NQ; ++q) {
      float c = cs[2 * q], s = cs[2 * q + 1];
      float nr = __builtin_fmaf(c, r[q], -s * im[q]);
      float ni = __builtin_fmaf(s, r[q],  c * im[q]);
      r[q] = nr; im[q] = ni;
    }
    // sequential entanglement scan (carry = previous qubit's new value)
    const float* st = ws + 2 * NL * NQ + L * (NQ - 1);
    float pr = r[0], pi = im[0];
    #pragma unroll
    for (int j = 0; j < NQ - 1; ++j) {
      float s  = st[j];
      float tr = r[j + 1], ti = im[j + 1];
      float amp = __builtin_amdgcn_sqrtf(__builtin_fmaf(pi, pi, pr * pr));
      float m = s * amp;
      pr = __builtin_fmaf(m, pr - tr, tr);   // (1-m)*tr + m*pr
      pi = __builtin_fmaf(m, pi - ti, ti);
      r[j + 1] = pr; im[j + 1] = pi;
    }
  }

  // measurement + normalization
  float meas[NQ];
  float sum = 1e-8f;
  #pragma unroll
  for (int q = 0; q < NQ; ++q) {
    meas[q] = __builtin_fmaf(im[q], im[q], r[q] * r[q]);
    sum += meas[q];
  }
  float inv = __builtin_amdgcn_rcpf(sum);

  float4* ov = (float4*)(out + (size_t)idx * NQ);
  #pragma unroll
  for (int k = 0; k < 4; ++k) {
    float4 v;
    v.x = meas[4 * k + 0] * inv;
    v.y = meas[4 * k + 1] * inv;
    v.z = meas[4 * k + 2] * inv;
    v.w = meas[4 * k + 3] * inv;
    ov[k] = v;
  }
}

extern "C" void kernel_launch(void* const* d_in, const int* in_sizes, int n_in,
                              void* d_out, int out_size, void* d_ws, size_t ws_size,
                              hipStream_t stream) {
  const float* x   = (const float*)d_in[0];   // (BATCH, 16) f32
  const float* rot = (const float*)d_in[1];   // (8, 16, 3) f32
  const float* ent = (const float*)d_in[2];   // (8, 15)    f32
  float* out = (float*)d_out;                 // (BATCH, 16) f32
  float* ws  = (float*)d_ws;                  // 376 floats used

  const int batch = in_sizes[0] / NQ;

  qc_prep<<<1, 128, 0, stream>>>(rot, ent, ws);

  const int grid = (batch + TPB - 1) / TPB;
  qc_main<<<grid, TPB, 0, stream>>>(x, ws, out, batch);
}